// ExpertChoiceRouter_62311385530872
// MI455X (gfx1250) — compile-verified
//
#include <hip/hip_runtime.h>
#include <math.h>

typedef float v2f __attribute__((ext_vector_type(2)));
typedef float v8f __attribute__((ext_vector_type(8)));

#define B_DIM 4
#define S_DIM 4096
#define H_DIM 2048
#define D_DIM 3
#define K_SEL 1365                  // int(4096/3)
#define T_TOK (B_DIM * S_DIM)       // 16384 tokens
#define N_TILE (T_TOK / 16)         // 1024 tiles of 16 tokens
#define WAVES_PER_BLOCK 8

// ---------------------------------------------------------------------------
// Kernel 1: scores via V_WMMA_F32_16X16X4_F32.
// One wave per 16-token tile. A = 16x4 f32 tile of hidden (16x4 layout:
// lanes 0-15 hold K=0,1 ; lanes 16-31 hold K=2,3). B = 4x16: columns 0..2 are
// routing rows, columns 3..15 read an all-zero LDS row (no mask multiply).
// rp is staged once per block into LDS (4 rows x 2048 f32 = 32 KB) so the
// inner loop issues only hidden global loads + broadcast ds loads + WMMA.
// ---------------------------------------------------------------------------
__global__ void __launch_bounds__(256)
router_scores_wmma(const float* __restrict__ hidden,
                   const float* __restrict__ rp,
                   float* __restrict__ scores0,   // [T_TOK]
                   float* __restrict__ partials)  // [N_TILE * 2 * D_DIM]
{
    __shared__ float rpl[4][H_DIM];    // rows 0..2 = routing params, row 3 = 0

    // Cooperative stage: 1536 float4 of rp + 512 float4 of zeros.
    {
        float* lf = &rpl[0][0];
        const float4* g4 = (const float4*)rp;
        float4* l4 = (float4*)lf;
        for (int i = threadIdx.x; i < (D_DIM * H_DIM) / 4; i += 256) l4[i] = g4[i];
        float4 z; z.x = 0.f; z.y = 0.f; z.z = 0.f; z.w = 0.f;
        float4* z4 = (float4*)(lf + D_DIM * H_DIM);
        for (int i = threadIdx.x; i < H_DIM / 4; i += 256) z4[i] = z;
    }
    __syncthreads();

    const int lane = threadIdx.x & 31;
    const int wave = threadIdx.x >> 5;
    const int tile = blockIdx.x * WAVES_PER_BLOCK + wave;
    const int base = tile * 16;            // first token of tile
    const int row  = lane & 15;            // A: matrix row M ; B/C/D: column N
    const int kh   = (lane >> 4) * 2;      // lanes 0-15 -> K 0,1 ; 16-31 -> K 2,3

    const float* __restrict__ Arow = hidden + (size_t)(base + row) * H_DIM + kh;
    const int   rclamp = (row < D_DIM) ? row : D_DIM;   // row 3 = zeros
    const float* Bl = &rpl[rclamp][kh];

    v8f c = {};
#pragma unroll 8
    for (int k0 = 0; k0 < H_DIM; k0 += 4) {
        v2f a, b;
        a.x = Arow[k0];
        a.y = Arow[k0 + 1];
        b = *(const v2f*)(Bl + k0);        // ds_load_b64, broadcast-friendly
        // 8 args: (neg_a, A, neg_b, B, c_mod, C, reuse_a, reuse_b)
        c = __builtin_amdgcn_wmma_f32_16x16x4_f32(
                false, a, false, b, (short)0, c, false, false);
    }

    // C/D layout: lane L (0-15): N=L, VGPR v -> M=v ; lane L (16-31): N=L-16, M=v+8
    if (row < D_DIM) {
        const int half = lane >> 4;   // 0: rows 0-7 ; 1: rows 8-15
        float s = 0.0f;
#pragma unroll
        for (int v = 0; v < 8; ++v) {
            const int tok = base + v + 8 * half;
            const float sc = __builtin_amdgcn_rcpf(1.0f + __expf(-c[v]));  // sigmoid
            if (row == 0) scores0[tok] = sc;
            s += __builtin_amdgcn_rcpf(1.0f + __expf(-sc));  // sigmoid(sigmoid)
        }
        partials[(size_t)(tile * 2 + half) * D_DIM + row] = s;
    }
}

// ---------------------------------------------------------------------------
// Kernel 2: per-batch exact top-k via binary search on f32 bit patterns
// (positive floats are monotone in their u32 bits). Then writes:
//   depth_out[b,s]   = selected ? 3 : 1   (step-0 winners end at depth 3)
//   masks[0][b,s]    = 1
//   masks[1/2][b,s]  = selected ? 1 : 0   (final active set)
// Ties at the threshold are taken by lowest index (top_k stable order).
// ---------------------------------------------------------------------------
__global__ void __launch_bounds__(256)
topk_select(const float* __restrict__ scores0,
            float* __restrict__ depth_out,   // [B,S]
            float* __restrict__ mask_out)    // [D,B,S]
{
    __shared__ float sv[S_DIM];
    __shared__ unsigned char ssel[S_DIM];
    __shared__ int cnt;

    const int b   = blockIdx.x;
    const int tid = threadIdx.x;
    const float* __restrict__ src = scores0 + (size_t)b * S_DIM;

    for (int i = tid; i < S_DIM; i += 256) sv[i] = src[i];
    __syncthreads();

    // invariant: count(v >= f(lo)) >= k ; count(v >= f(hi)) < k
    unsigned lo = 0u, hi = 0x3F800000u;   // [0.0, 1.0); sigmoid in (0,1)
    while (hi - lo > 1u) {
        const unsigned mid = lo + (hi - lo) / 2u;
        const float t = __uint_as_float(mid);
        if (tid == 0) cnt = 0;
        __syncthreads();
        int local = 0;
        for (int i = tid; i < S_DIM; i += 256) local += (sv[i] >= t) ? 1 : 0;
        atomicAdd(&cnt, local);
        __syncthreads();
        const int cge = cnt;
        __syncthreads();
        if (cge >= K_SEL) lo = mid; else hi = mid;
    }
    const float T = __uint_as_float(lo);   // exact k-th largest value

    // strictly-greater count (ties handled separately, stable by index)
    if (tid == 0) cnt = 0;
    __syncthreads();
    {
        int local = 0;
        for (int i = tid; i < S_DIM; i += 256) local += (sv[i] > T) ? 1 : 0;
        atomicAdd(&cnt, local);
    }
    __syncthreads();
    const int cgt = cnt;

    for (int i = tid; i < S_DIM; i += 256) ssel[i] = (sv[i] > T) ? 1 : 0;
    __syncthreads();
    if (tid == 0) {                        // take lowest-index ties up to k
        int need = K_SEL - cgt;
        for (int i = 0; i < S_DIM && need > 0; ++i) {
            if (sv[i] == T && ssel[i] == 0) { ssel[i] = 1; --need; }
        }
    }
    __syncthreads();

    for (int i = tid; i < S_DIM; i += 256) {
        const float on = ssel[i] ? 1.0f : 0.0f;
        const size_t p = (size_t)b * S_DIM + i;
        depth_out[p]              = ssel[i] ? 3.0f : 1.0f;
        mask_out[0 * T_TOK + p]   = 1.0f;
        mask_out[1 * T_TOK + p]   = on;
        mask_out[2 * T_TOK + p]   = on;
    }
}

// ---------------------------------------------------------------------------
// Kernel 3: deterministic fixed-order reduction of partials -> KL loss.
// ---------------------------------------------------------------------------
__global__ void finalize_loss(const float* __restrict__ partials,
                              float* __restrict__ loss_out)
{
    if (threadIdx.x == 0 && blockIdx.x == 0) {
        float sums[D_DIM] = {0.0f, 0.0f, 0.0f};
        for (int t = 0; t < N_TILE * 2; ++t)
            for (int r = 0; r < D_DIM; ++r)
                sums[r] += partials[(size_t)t * D_DIM + r];
        const float target = 1.0f / (float)D_DIM;
        const float logt   = logf(target);
        float l = 0.0f;
        for (int r = 0; r < D_DIM; ++r) {
            const float p = sums[r] / (float)T_TOK;
            l += target * (logt - logf(p));
        }
        loss_out[0] = l / (float)D_DIM;    // batchmean with input size D
    }
}

extern "C" void kernel_launch(void* const* d_in, const int* in_sizes, int n_in,
                              void* d_out, int out_size, void* d_ws, size_t ws_size,
                              hipStream_t stream) {
    (void)in_sizes; (void)n_in; (void)out_size; (void)ws_size;
    const float* hidden = (const float*)d_in[0];   // [4,4096,2048] f32
    const float* rp     = (const float*)d_in[1];   // [3,2048] f32

    float* out       = (float*)d_out;
    float* depth_out = out;                        // 16384 floats
    float* loss_out  = out + T_TOK;                // 1 float
    float* mask_out  = out + T_TOK + 1;            // 3*16384 floats

    float* scores0  = (float*)d_ws;                // 16384 floats (overwritten each call)
    float* partials = scores0 + T_TOK;             // 1024*2*3 floats (overwritten each call)

    router_scores_wmma<<<N_TILE / WAVES_PER_BLOCK, 32 * WAVES_PER_BLOCK, 0, stream>>>(
        hidden, rp, scores0, partials);
    topk_select<<<B_DIM, 256, 0, stream>>>(scores0, depth_out, mask_out);
    finalize_loss<<<1, 32, 0, stream>>>(partials, loss_out);
}